// CowDetectionPredictor_72430328479813
// MI455X (gfx1250) — compile-verified
//
#include <hip/hip_runtime.h>
#include <hip/hip_bf16.h>
#include <math.h>

#define IMG   4320
#define HALFI 2160
#define RED   1080
#define TT    10      // reduced-tile side per workgroup
#define NGRID 108     // 1080 / TT

typedef __attribute__((ext_vector_type(2))) float v2f;
typedef __attribute__((ext_vector_type(8))) float v8f;
typedef __attribute__((ext_vector_type(2))) int   v2i;

#if __has_builtin(__builtin_amdgcn_global_load_async_to_lds_b64) && \
    __has_builtin(__builtin_amdgcn_s_wait_asynccnt)
#define ATHENA_ASYNC 1
typedef v2i __attribute__((address_space(1)))* gv2i_t;   // global-AS vector ptr
typedef v2i __attribute__((address_space(3)))* lv2i_t;   // LDS-AS vector ptr
#endif

// 8-byte global->LDS copy; async (ASYNCcnt / no VGPR round trip) when available.
__device__ __forceinline__ void ld2(const float* g, float* l) {
#ifdef ATHENA_ASYNC
  __builtin_amdgcn_global_load_async_to_lds_b64((gv2i_t)g, (lv2i_t)l, 0, 0);
#else
  *(v2f*)l = *(const v2f*)g;
#endif
}

// ---------------------------------------------------------------------------
// Kernel 1: fully fused backbone.
// conv1(3->3) -> conv2(3->2) -> maxpool2 -> conv3(2->2) -> conv4(2->1) -> maxpool2
// One workgroup computes a 10x10 tile of the 1080x1080 "reduced" map; the 52x52x3
// x-patch and all intermediates live in LDS (ping-pong). Weights are read via
// uniform addresses (SGPRs); activations via ds_load_b64, 2 output cols/thread.
// ---------------------------------------------------------------------------
__global__ __launch_bounds__(256) void backbone_fused(
    const float* __restrict__ x,
    const float* __restrict__ w1, const float* __restrict__ b1,
    const float* __restrict__ w2, const float* __restrict__ b2,
    const float* __restrict__ w3, const float* __restrict__ b3,
    const float* __restrict__ w4, const float* __restrict__ b4,
    float* __restrict__ red) {
  __shared__ __align__(16) float bufA[3 * 52 * 52];   // x tile -> c2 -> c3
  __shared__ __align__(16) float bufB[3 * 50 * 50];   // c1 -> p1 -> c4
  const int tid = threadIdx.x;
  const int R0 = blockIdx.y * TT, C0 = blockIdx.x * TT;
  const int xr0 = 4 * R0 - 6, xc0 = 4 * C0 - 6;

  // ---- stage x tile: 3 x 52 x 52 ----
  const bool interior = (xr0 >= 0) && (xr0 + 52 <= IMG) &&
                        (xc0 >= 0) && (xc0 + 52 <= IMG);
  if (interior) {
    for (int idx = tid; idx < 3 * 52 * 26; idx += 256) {
      int ci = idx / 1352, rem = idx % 1352, r = rem / 26, cp = 2 * (rem % 26);
      const float* g = x + (size_t)ci * IMG * IMG + (size_t)(xr0 + r) * IMG + (xc0 + cp);
      ld2(g, bufA + ci * 2704 + r * 52 + cp);
    }
#ifdef ATHENA_ASYNC
    __builtin_amdgcn_s_wait_asynccnt(0);
#endif
  } else {
    for (int idx = tid; idx < 3 * 52 * 52; idx += 256) {
      int ci = idx / 2704, rem = idx % 2704, r = rem / 52, c = rem % 52;
      int gr = xr0 + r, gc = xc0 + c;
      float v = 0.f;
      if (gr >= 0 && gr < IMG && gc >= 0 && gc < IMG)
        v = x[(size_t)ci * IMG * IMG + (size_t)gr * IMG + gc];
      bufA[idx] = v;
    }
  }
  __syncthreads();

  // ---- conv1: bufB[co][50][50], zero where output outside image ----
  const int c1r0 = 4 * R0 - 5, c1c0 = 4 * C0 - 5;
  for (int co = 0; co < 3; ++co) {
    float wr[27];
#pragma unroll
    for (int k = 0; k < 27; ++k) wr[k] = w1[co * 27 + k];
    const float bias = b1[co];
    for (int t = tid; t < 50 * 25; t += 256) {
      int r = t / 25, c = (t % 25) * 2;
      float a0 = bias, a1 = bias;
#pragma unroll
      for (int ci = 0; ci < 3; ++ci)
#pragma unroll
        for (int ky = 0; ky < 3; ++ky) {
          const v2f* row = (const v2f*)(bufA + ci * 2704 + (r + ky) * 52 + c);
          v2f p0 = row[0], p1 = row[1];
          float u0 = wr[ci * 9 + ky * 3 + 0];
          float u1 = wr[ci * 9 + ky * 3 + 1];
          float u2 = wr[ci * 9 + ky * 3 + 2];
          a0 += u0 * p0.x + u1 * p0.y + u2 * p1.x;
          a1 += u0 * p0.y + u1 * p1.x + u2 * p1.y;
        }
      int gr = c1r0 + r, gc = c1c0 + c;
      bool vr = (gr >= 0) && (gr < IMG);
      bufB[co * 2500 + r * 50 + c]     = (vr && gc >= 0 && gc < IMG)         ? a0 : 0.f;
      bufB[co * 2500 + r * 50 + c + 1] = (vr && gc + 1 >= 0 && gc + 1 < IMG) ? a1 : 0.f;
    }
  }
  __syncthreads();

  // ---- conv2: bufA[co][48][48] ----
  for (int co = 0; co < 2; ++co) {
    float wr[27];
#pragma unroll
    for (int k = 0; k < 27; ++k) wr[k] = w2[co * 27 + k];
    const float bias = b2[co];
    for (int t = tid; t < 48 * 24; t += 256) {
      int r = t / 24, c = (t % 24) * 2;
      float a0 = bias, a1 = bias;
#pragma unroll
      for (int ci = 0; ci < 3; ++ci)
#pragma unroll
        for (int ky = 0; ky < 3; ++ky) {
          const v2f* row = (const v2f*)(bufB + ci * 2500 + (r + ky) * 50 + c);
          v2f p0 = row[0], p1 = row[1];
          float u0 = wr[ci * 9 + ky * 3 + 0];
          float u1 = wr[ci * 9 + ky * 3 + 1];
          float u2 = wr[ci * 9 + ky * 3 + 2];
          a0 += u0 * p0.x + u1 * p0.y + u2 * p1.x;
          a1 += u0 * p0.y + u1 * p1.x + u2 * p1.y;
        }
      bufA[co * 2304 + r * 48 + c]     = a0;
      bufA[co * 2304 + r * 48 + c + 1] = a1;
    }
  }
  __syncthreads();

  // ---- pool1: bufB[c][24][24], zero where pooled row/col outside 2160 map ----
  const int p1r0 = 2 * R0 - 2, p1c0 = 2 * C0 - 2;
  for (int idx = tid; idx < 2 * 576; idx += 256) {
    int ch = idx / 576, rem = idx % 576, r = rem / 24, c = rem % 24;
    int u = p1r0 + r, v = p1c0 + c;
    float val = 0.f;
    if (u >= 0 && u < HALFI && v >= 0 && v < HALFI) {
      v2f q0 = *(const v2f*)(bufA + ch * 2304 + (2 * r) * 48 + 2 * c);
      v2f q1 = *(const v2f*)(bufA + ch * 2304 + (2 * r + 1) * 48 + 2 * c);
      val = fmaxf(fmaxf(q0.x, q0.y), fmaxf(q1.x, q1.y));
    }
    bufB[idx] = val;
  }
  __syncthreads();

  // ---- conv3: bufA[co][22][22], zero where outside 2160 map ----
  const int c3r0 = 2 * R0 - 1, c3c0 = 2 * C0 - 1;
  for (int co = 0; co < 2; ++co) {
    float wr[18];
#pragma unroll
    for (int k = 0; k < 18; ++k) wr[k] = w3[co * 18 + k];
    const float bias = b3[co];
    for (int t = tid; t < 22 * 11; t += 256) {
      int r = t / 11, c = (t % 11) * 2;
      float a0 = bias, a1 = bias;
#pragma unroll
      for (int ci = 0; ci < 2; ++ci)
#pragma unroll
        for (int ky = 0; ky < 3; ++ky) {
          const v2f* row = (const v2f*)(bufB + ci * 576 + (r + ky) * 24 + c);
          v2f p0 = row[0], p1 = row[1];
          float u0 = wr[ci * 9 + ky * 3 + 0];
          float u1 = wr[ci * 9 + ky * 3 + 1];
          float u2 = wr[ci * 9 + ky * 3 + 2];
          a0 += u0 * p0.x + u1 * p0.y + u2 * p1.x;
          a1 += u0 * p0.y + u1 * p1.x + u2 * p1.y;
        }
      int gr = c3r0 + r, gc = c3c0 + c;
      bool vr = (gr >= 0) && (gr < HALFI);
      bufA[co * 484 + r * 22 + c]     = (vr && gc >= 0 && gc < HALFI)         ? a0 : 0.f;
      bufA[co * 484 + r * 22 + c + 1] = (vr && gc + 1 >= 0 && gc + 1 < HALFI) ? a1 : 0.f;
    }
  }
  __syncthreads();

  // ---- conv4: bufB[20][20] ----
  {
    float wr[18];
#pragma unroll
    for (int k = 0; k < 18; ++k) wr[k] = w4[k];
    const float bias = b4[0];
    for (int t = tid; t < 20 * 10; t += 256) {
      int r = t / 10, c = (t % 10) * 2;
      float a0 = bias, a1 = bias;
#pragma unroll
      for (int ci = 0; ci < 2; ++ci)
#pragma unroll
        for (int ky = 0; ky < 3; ++ky) {
          const v2f* row = (const v2f*)(bufA + ci * 484 + (r + ky) * 22 + c);
          v2f p0 = row[0], p1 = row[1];
          float u0 = wr[ci * 9 + ky * 3 + 0];
          float u1 = wr[ci * 9 + ky * 3 + 1];
          float u2 = wr[ci * 9 + ky * 3 + 2];
          a0 += u0 * p0.x + u1 * p0.y + u2 * p1.x;
          a1 += u0 * p0.y + u1 * p1.x + u2 * p1.y;
        }
      bufB[r * 20 + c]     = a0;
      bufB[r * 20 + c + 1] = a1;
    }
  }
  __syncthreads();

  // ---- pool2 + store reduced tile ----
  for (int idx = tid; idx < 100; idx += 256) {
    int r = idx / 10, c = idx % 10;
    v2f q0 = *(const v2f*)(bufB + (2 * r) * 20 + 2 * c);
    v2f q1 = *(const v2f*)(bufB + (2 * r + 1) * 20 + 2 * c);
    red[(size_t)(R0 + r) * RED + (C0 + c)] =
        fmaxf(fmaxf(q0.x, q0.y), fmaxf(q1.x, q1.y));
  }
}

// ---------------------------------------------------------------------------
// Kernel 2: RPN 50x50 stride-50 conv as a WMMA GEMM.
// y[n, co] = b[co] + sum_k W[co,k] * patch[n,k];  M=441(+pad) N=2(+pad) K=2500.
// One wave per 16-row M tile; V_WMMA_F32_16X16X4_F32, 625 K-steps.
// A layout (16x4 f32): lanes 0-15 hold K={0,1}, lanes 16-31 hold K={2,3}.
// ---------------------------------------------------------------------------
__global__ __launch_bounds__(32) void rpn_conv_wmma(
    const float* __restrict__ red, const float* __restrict__ w,
    const float* __restrict__ b, float* __restrict__ y) {
  const int lane = threadIdx.x;
  const int half = lane >> 4;       // 0 -> K slice {0,1}; 1 -> {2,3}
  const int nl   = lane & 15;       // N column / M row within half
  const int m    = blockIdx.x * 16 + nl;
  const bool valid = (m < 441);
  const int pi = valid ? (m / 21) : 0;
  const int pj = valid ? (m % 21) : 0;
  const float* base = red + (size_t)(pi * 50) * RED + pj * 50;

  v8f acc = {};
  for (int k = 0; k < 2500; k += 4) {
    int k0 = k + half * 2;
    int u0 = k0 / 50,       v0 = k0 % 50;
    int u1 = (k0 + 1) / 50, v1 = (k0 + 1) % 50;
    v2f a;
    a.x = valid ? base[u0 * RED + v0] : 0.f;
    a.y = valid ? base[u1 * RED + v1] : 0.f;
    v2f bb;
    bb.x = (nl < 2) ? w[nl * 2500 + k0]     : 0.f;
    bb.y = (nl < 2) ? w[nl * 2500 + k0 + 1] : 0.f;
    acc = __builtin_amdgcn_wmma_f32_16x16x4_f32(
        /*neg_a=*/false, a, /*neg_b=*/false, bb,
        /*c_mod=*/(short)0, acc, /*reuse_a=*/false, /*reuse_b=*/false);
  }
  // D layout: VGPR r holds M=r (lanes 0-15) / M=r+8 (lanes 16-31), N = nl.
  if (nl < 2) {
    float bias = b[nl];
    for (int r = 0; r < 8; ++r) {
      int mm = blockIdx.x * 16 + r + half * 8;
      if (mm < 441) y[nl * 441 + mm] = acc[r] + bias;
    }
  }
}

// ---------------------------------------------------------------------------
// Kernel 3: rpn_bbox (24ch) and rpn_cls (6ch, tanh) 3x3 pad-1 convs on (2,21,21)
// ---------------------------------------------------------------------------
__global__ __launch_bounds__(256) void rpn_heads(
    const float* __restrict__ y,
    const float* __restrict__ wb, const float* __restrict__ bb,
    const float* __restrict__ wc, const float* __restrict__ bc,
    float* __restrict__ offsets, float* __restrict__ obj) {
  __shared__ float ys[2 * 441];
  for (int i = threadIdx.x; i < 882; i += 256) ys[i] = y[i];
  __syncthreads();
  for (int idx = threadIdx.x; idx < 30 * 441; idx += 256) {
    int o = idx / 441, p = idx % 441, i = p / 21, j = p % 21;
    const float* w; float bias; int oc;
    if (o < 24) { oc = o;      w = wb + oc * 18; bias = bb[oc]; }
    else        { oc = o - 24; w = wc + oc * 18; bias = bc[oc]; }
    float acc = bias;
    for (int ci = 0; ci < 2; ++ci)
      for (int ky = 0; ky < 3; ++ky) {
        int r = i + ky - 1; if (r < 0 || r >= 21) continue;
        for (int kx = 0; kx < 3; ++kx) {
          int c = j + kx - 1; if (c < 0 || c >= 21) continue;
          acc += w[ci * 9 + ky * 3 + kx] * ys[ci * 441 + r * 21 + c];
        }
      }
    if (o < 24) offsets[o * 441 + p] = acc;
    else        obj[oc * 441 + p]    = tanhf(acc);
  }
}

// ---------------------------------------------------------------------------
// Kernel 4: selection (first 20 Trues of broadcast mask, row-major) + proposals.
// Faithful to reference: apply_offsets called with SWAPPED args (a=offsets,
// o=anchors), clip to [0,21], scale by 50.
// ---------------------------------------------------------------------------
__global__ void select_props(const float* __restrict__ offsets,
                             const float* __restrict__ obj,
                             float* __restrict__ props) {
  if (threadIdx.x != 0) return;
  int sel[20]; int cnt = 0;
  for (int a = 0; a < 6 && cnt < 20; ++a)
    for (int c = 0; c < 4 && cnt < 20; ++c)
      for (int p = 0; p < 441 && cnt < 20; ++p)
        if (obj[a * 441 + p] > 0.95f) sel[cnt++] = a * 1764 + c * 441 + p;
  // argsort(~mask, stable) semantics: Falses follow, ascending.
  for (int a = 0; a < 6 && cnt < 20; ++a)
    for (int c = 0; c < 4 && cnt < 20; ++c)
      for (int p = 0; p < 441 && cnt < 20; ++p)
        if (!(obj[a * 441 + p] > 0.95f)) sel[cnt++] = a * 1764 + c * 441 + p;

  for (int k = 0; k < 5; ++k) {
    float av[4], ov[4];
    for (int q = 0; q < 4; ++q) {
      int s = sel[k * 4 + q];
      int a = s / 1764, rem = s % 1764, c = rem / 441, p = rem % 441;
      int i = p / 21, j = p % 21;
      float sz = (a < 3) ? 1.f : 2.f;
      int am = a % 3;
      float ar = (am == 0) ? 0.5f : (am == 1 ? 1.f : 2.f);
      float bse;
      if (c == 0)      bse = 0.f;
      else if (c == 1) bse = sz * -(ar - 1.f) * 0.5f;
      else if (c == 2) bse = sz;
      else             bse = sz * (1.f + (ar - 1.f) * 0.5f);
      av[q] = bse + (((c & 1) == 0) ? (float)i : (float)j);
      ov[q] = offsets[s];
    }
    float r0 = ov[0] + av[0] - av[2] * 0.5f;
    float r1 = ov[1] - av[3] * 0.5f;
    float r2 = ov[2] + av[0] + av[2] * 0.5f;
    float r3 = ov[3] + av[3] * 0.5f;
    props[k * 4 + 0] = fminf(fmaxf(r0, 0.f), 21.f) * 50.f;
    props[k * 4 + 1] = fminf(fmaxf(r1, 0.f), 21.f) * 50.f;
    props[k * 4 + 2] = fminf(fmaxf(r2, 0.f), 21.f) * 50.f;
    props[k * 4 + 3] = fminf(fmaxf(r3, 0.f), 21.f) * 50.f;
  }
}

// ---------------------------------------------------------------------------
// Kernel 5: torchvision-style quantized 2x2 RoIPool; one block per (roi,cell).
// ---------------------------------------------------------------------------
__global__ __launch_bounds__(256) void roipool(
    const float* __restrict__ red, const float* __restrict__ props,
    float* __restrict__ pooled) {
  const int k = blockIdx.x >> 2, cell = blockIdx.x & 3;
  const int ph = cell >> 1, pw = cell & 1;
  float x1 = rintf(props[k * 4 + 0]), y1 = rintf(props[k * 4 + 1]);
  float x2 = rintf(props[k * 4 + 2]), y2 = rintf(props[k * 4 + 3]);
  float bw = fmaxf(x2 - x1 + 1.f, 1.f) * 0.5f;
  float bh = fmaxf(y2 - y1 + 1.f, 1.f) * 0.5f;
  float fws = fminf(fmaxf(floorf((float)pw * bw) + x1, 0.f), 1080.f);
  float fwe = fminf(fmaxf(ceilf((float)(pw + 1) * bw) + x1, 0.f), 1080.f);
  float fhs = fminf(fmaxf(floorf((float)ph * bh) + y1, 0.f), 1080.f);
  float fhe = fminf(fmaxf(ceilf((float)(ph + 1) * bh) + y1, 0.f), 1080.f);
  int ws = (int)fws, we = (int)fwe, hs = (int)fhs, he = (int)fhe;
  int nw = we - ws;
  int total = (he - hs) * nw;
  float m = -3.402823466e38f;
  for (int t = threadIdx.x; t < total; t += 256) {
    int r = hs + t / nw, c = ws + t % nw;
    m = fmaxf(m, red[(size_t)r * RED + c]);
  }
  __shared__ float sm[256];
  sm[threadIdx.x] = m;
  __syncthreads();
  for (int s = 128; s > 0; s >>= 1) {
    if (threadIdx.x < s) sm[threadIdx.x] = fmaxf(sm[threadIdx.x], sm[threadIdx.x + s]);
    __syncthreads();
  }
  if (threadIdx.x == 0)
    pooled[k * 4 + ph * 2 + pw] = (total > 0) ? sm[0] : 0.f;  // empty -> 0
}

// ---------------------------------------------------------------------------
// Kernel 6: FC heads, final box offsets (a=props, o=box_off), clip to
// (h=1, w=3) as in the reference, softmax over 2 classes.
// ---------------------------------------------------------------------------
__global__ void heads(const float* __restrict__ props,
                      const float* __restrict__ pooled,
                      const float* __restrict__ fc_w, const float* __restrict__ fc_b,
                      const float* __restrict__ box_w, const float* __restrict__ box_b,
                      const float* __restrict__ cls_w, const float* __restrict__ cls_b,
                      float* __restrict__ out) {
  if (threadIdx.x != 0) return;
  float fc[5][12];
  for (int k = 0; k < 5; ++k)
    for (int n = 0; n < 12; ++n) {
      float acc = fc_b[n];
      for (int q = 0; q < 4; ++q) acc += pooled[k * 4 + q] * fc_w[n * 4 + q];
      fc[k][n] = acc;
    }
  for (int k = 0; k < 5; ++k) {
    float bo[4];
    for (int n = 0; n < 4; ++n) {
      float acc = box_b[n];
      for (int q = 0; q < 12; ++q) acc += fc[k][q] * box_w[n * 12 + q];
      bo[n] = acc;
    }
    float a0 = props[k * 4 + 0], a1 = props[k * 4 + 1];
    float a2 = props[k * 4 + 2], a3 = props[k * 4 + 3];
    float r0 = a0 + bo[0] - bo[2] * 0.5f;
    float r1 = a1 - bo[3] * 0.5f;
    float r2 = a2 + bo[0] + bo[2] * 0.5f;
    float r3 = a3 + bo[3] * 0.5f;
    out[k * 4 + 0] = fminf(fmaxf(r0, 0.f), 3.f);
    out[k * 4 + 1] = fminf(fmaxf(r1, 0.f), 1.f);
    out[k * 4 + 2] = fminf(fmaxf(r2, 0.f), 3.f);
    out[k * 4 + 3] = fminf(fmaxf(r3, 0.f), 1.f);
    float l0 = cls_b[0], l1 = cls_b[1];
    for (int q = 0; q < 12; ++q) {
      l0 += fc[k][q] * cls_w[q];
      l1 += fc[k][q] * cls_w[12 + q];
    }
    float mx = fmaxf(l0, l1);
    float e0 = expf(l0 - mx), e1 = expf(l1 - mx);
    float s = e0 + e1;
    out[20 + k * 2 + 0] = e0 / s;
    out[20 + k * 2 + 1] = e1 / s;
  }
}

// ---------------------------------------------------------------------------
extern "C" void kernel_launch(void* const* d_in, const int* in_sizes, int n_in,
                              void* d_out, int out_size, void* d_ws, size_t ws_size,
                              hipStream_t stream) {
  (void)in_sizes; (void)n_in; (void)out_size; (void)ws_size;
  const float* x      = (const float*)d_in[0];
  const float* w1     = (const float*)d_in[1];
  const float* b1     = (const float*)d_in[2];
  const float* w2     = (const float*)d_in[3];
  const float* b2     = (const float*)d_in[4];
  const float* w3     = (const float*)d_in[5];
  const float* b3     = (const float*)d_in[6];
  const float* w4     = (const float*)d_in[7];
  const float* b4     = (const float*)d_in[8];
  const float* rpnw   = (const float*)d_in[9];
  const float* rpnb   = (const float*)d_in[10];
  const float* bbw    = (const float*)d_in[11];
  const float* bbb    = (const float*)d_in[12];
  const float* clw    = (const float*)d_in[13];
  const float* clb    = (const float*)d_in[14];
  const float* fcw    = (const float*)d_in[15];
  const float* fcb    = (const float*)d_in[16];
  const float* boxw   = (const float*)d_in[17];
  const float* boxb   = (const float*)d_in[18];
  const float* clsw   = (const float*)d_in[19];
  const float* clsb   = (const float*)d_in[20];

  float* ws     = (float*)d_ws;
  float* red    = ws;                    // 1080*1080
  float* y      = red + 1166400;         // 2*441
  float* offs   = y + 882;               // 24*441
  float* obj    = offs + 10584;          // 6*441
  float* props  = obj + 2646;            // 5*4
  float* pooled = props + 20;            // 5*4
  float* out    = (float*)d_out;         // 20 boxes + 10 cls

  backbone_fused<<<dim3(NGRID, NGRID), 256, 0, stream>>>(
      x, w1, b1, w2, b2, w3, b3, w4, b4, red);
  rpn_conv_wmma<<<28, 32, 0, stream>>>(red, rpnw, rpnb, y);
  rpn_heads<<<1, 256, 0, stream>>>(y, bbw, bbb, clw, clb, offs, obj);
  select_props<<<1, 32, 0, stream>>>(offs, obj, props);
  roipool<<<20, 256, 0, stream>>>(red, props, pooled);
  heads<<<1, 32, 0, stream>>>(props, pooled, fcw, fcb, boxw, boxb, clsw, clsb, out);
}